// GraphSAGEEncoder_59030030516960
// MI455X (gfx1250) — compile-verified
//
#include <hip/hip_runtime.h>
#include <hip/hip_bf16.h>

// GraphSAGE encoder for MI455X (gfx1250, wave32).
// All three GEMMs use V_WMMA_F32_16X16X4_F32 (fp32 matrix pipe, matches the
// fp32 reference numerics). Working set after the 51 MB feat read (h, agg,
// deg ~ 26 MB) is resident in the 192 MB L2; segment_max is implemented as
// u32 atomicMax on float bit patterns (valid: sigmoid outputs > 0, init 0).

typedef __attribute__((ext_vector_type(2))) float v2f;
typedef __attribute__((ext_vector_type(8))) float v8f;

#define D_IN 128
#define D_H  32

__device__ __forceinline__ v8f wmma_f32x4(v2f a, v2f b, v8f c) {
  // 8-arg form: (neg_a, A, neg_b, B, c_mod, C, reuse_a, reuse_b)
  return __builtin_amdgcn_wmma_f32_16x16x4_f32(false, a, false, b,
                                               (short)0, c, false, false);
}

// ---------------------------------------------------------------- h = feat@W_in + b_in
__global__ __launch_bounds__(256) void k_input_proj(
    const float* __restrict__ feat, const float* __restrict__ W_in,
    const float* __restrict__ b_in, float* __restrict__ h, int n) {
  const int lane = threadIdx.x & 31;
  const int wv   = threadIdx.x >> 5;
  const int rowBase = (blockIdx.x * 8 + wv) * 16;   // uniform per wave
  if (rowBase >= n) return;
  int m = rowBase + (lane & 15);
  if (m >= n) m = n - 1;                            // clamp; stores guarded below
  const int kOff = (lane >> 4) * 2;
  const int nlo  = lane & 15;
  const float* arow = feat + (size_t)m * D_IN;
  v8f acc0 = {}; v8f acc1 = {};
#pragma unroll
  for (int k = 0; k < D_IN; k += 4) {
    const int kk = k + kOff;
    v2f a = *(const v2f*)(arow + kk);
    v2f b0, b1;
    b0.x = W_in[kk * D_H + nlo];        b0.y = W_in[(kk + 1) * D_H + nlo];
    b1.x = W_in[kk * D_H + 16 + nlo];   b1.y = W_in[(kk + 1) * D_H + 16 + nlo];
    acc0 = wmma_f32x4(a, b0, acc0);
    acc1 = wmma_f32x4(a, b1, acc1);
  }
  const int rOff = (lane >> 4) * 8;
#pragma unroll
  for (int j = 0; j < 8; ++j) {
    const int row = rowBase + rOff + j;
    if (row < n) {
      h[(size_t)row * D_H + nlo]      = acc0[j] + b_in[nlo];
      h[(size_t)row * D_H + 16 + nlo] = acc1[j] + b_in[16 + nlo];
    }
  }
}

// ---------------------------------------------------------------- zero agg / deg
__global__ void k_init(float* __restrict__ agg, int* __restrict__ deg, int n) {
  const int i = blockIdx.x * blockDim.x + threadIdx.x;
  if (i < n * D_H) agg[i] = 0.0f;
  if (i < n) deg[i] = 0;
}

// ---------------------------------------------------------------- degree counts
__global__ void k_deg(const long long* __restrict__ src,
                      const long long* __restrict__ dst,
                      int* __restrict__ deg, int e) {
  const int i = blockIdx.x * blockDim.x + threadIdx.x;
  if (i >= 2 * e) return;
  const long long seg = (i < e) ? dst[i] : src[i - e];
  atomicAdd(&deg[(int)seg], 1);
}

// ------------------------------------- t = sigmoid(h[nbr]@W_nbr + b); agg = segmax(t)
__global__ __launch_bounds__(256) void k_nbr(
    const float* __restrict__ h, const long long* __restrict__ src,
    const long long* __restrict__ dst, const float* __restrict__ W_nbr,
    const float* __restrict__ b_nbr, const float* __restrict__ b_n1,
    float* __restrict__ agg, int e) {
  const int lane = threadIdx.x & 31;
  const int wv   = threadIdx.x >> 5;
  const long long total = 2LL * e;
  const long long base  = ((long long)blockIdx.x * 8 + wv) * 16;  // uniform per wave
  if (base >= total) return;
  long long i = base + (lane & 15);
  if (i >= total) i = total - 1;                    // dup work; scatter guarded below
  const long long nbr = (i < e) ? src[i] : dst[i - e];
  const int kOff = (lane >> 4) * 2;
  const int nlo  = lane & 15;
  const float* arow = h + (size_t)nbr * D_H;        // L2-resident gather
  v8f acc0 = {}; v8f acc1 = {};
#pragma unroll
  for (int k = 0; k < D_H; k += 4) {
    const int kk = k + kOff;
    v2f a = *(const v2f*)(arow + kk);
    v2f b0, b1;
    b0.x = W_nbr[kk * D_H + nlo];       b0.y = W_nbr[(kk + 1) * D_H + nlo];
    b1.x = W_nbr[kk * D_H + 16 + nlo];  b1.y = W_nbr[(kk + 1) * D_H + 16 + nlo];
    acc0 = wmma_f32x4(a, b0, acc0);
    acc1 = wmma_f32x4(a, b1, acc1);
  }
  const float bia0 = b_nbr[nlo] + b_n1[nlo];
  const float bia1 = b_nbr[16 + nlo] + b_n1[16 + nlo];
  const int rOff = (lane >> 4) * 8;
#pragma unroll
  for (int j = 0; j < 8; ++j) {
    const long long i2 = base + rOff + j;
    if (i2 >= total) continue;
    const long long seg = (i2 < e) ? dst[i2] : src[i2 - e];
    float y0 = 1.0f / (1.0f + expf(-(acc0[j] + bia0)));
    float y1 = 1.0f / (1.0f + expf(-(acc1[j] + bia1)));
    // sigmoid > 0 and agg init 0: u32 ordering == float ordering for >= 0
    unsigned int* ap = (unsigned int*)(agg + (size_t)seg * D_H);
    atomicMax(ap + nlo,      __float_as_uint(y0));
    atomicMax(ap + 16 + nlo, __float_as_uint(y1));
  }
}

// ---------------------------------------- out = [h, mask(agg)] @ W_ffnn + b_ffnn
__global__ __launch_bounds__(256) void k_ffnn(
    const float* __restrict__ h, const float* __restrict__ agg,
    const int* __restrict__ deg, const float* __restrict__ W_ffnn,
    const float* __restrict__ b_ffnn, float* __restrict__ out, int n) {
  const int lane = threadIdx.x & 31;
  const int wv   = threadIdx.x >> 5;
  const int rowBase = (blockIdx.x * 8 + wv) * 16;   // uniform per wave
  if (rowBase >= n) return;
  int m = rowBase + (lane & 15);
  if (m >= n) m = n - 1;
  const int kOff = (lane >> 4) * 2;
  const int nlo  = lane & 15;
  const float* hrow = h   + (size_t)m * D_H;
  const float* grow = agg + (size_t)m * D_H;
  const float msk = (deg[m] > 1) ? 1.0f : 0.0f;     // per-element A mask, no EXEC change
  v8f acc0 = {}; v8f acc1 = {};
#pragma unroll
  for (int k = 0; k < 2 * D_H; k += 4) {            // kk never straddles the halves
    const int kk = k + kOff;
    v2f a;
    if (kk < D_H) {
      a = *(const v2f*)(hrow + kk);
    } else {
      v2f g = *(const v2f*)(grow + (kk - D_H));
      a.x = g.x * msk; a.y = g.y * msk;
    }
    v2f b0, b1;
    b0.x = W_ffnn[kk * D_H + nlo];       b0.y = W_ffnn[(kk + 1) * D_H + nlo];
    b1.x = W_ffnn[kk * D_H + 16 + nlo];  b1.y = W_ffnn[(kk + 1) * D_H + 16 + nlo];
    acc0 = wmma_f32x4(a, b0, acc0);
    acc1 = wmma_f32x4(a, b1, acc1);
  }
  const int rOff = (lane >> 4) * 8;
#pragma unroll
  for (int j = 0; j < 8; ++j) {
    const int row = rowBase + rOff + j;
    if (row < n) {
      out[(size_t)row * D_H + nlo]      = acc0[j] + b_ffnn[nlo];
      out[(size_t)row * D_H + 16 + nlo] = acc1[j] + b_ffnn[16 + nlo];
    }
  }
}

extern "C" void kernel_launch(void* const* d_in, const int* in_sizes, int n_in,
                              void* d_out, int out_size, void* d_ws, size_t ws_size,
                              hipStream_t stream) {
  const float*     feat   = (const float*)d_in[0];
  const long long* src    = (const long long*)d_in[1];  // int64 per reference
  const long long* dst    = (const long long*)d_in[2];
  const float*     W_in   = (const float*)d_in[3];
  const float*     b_in   = (const float*)d_in[4];
  const float*     W_nbr  = (const float*)d_in[5];
  const float*     b_nbr  = (const float*)d_in[6];
  const float*     b_n1   = (const float*)d_in[7];
  const float*     W_ffnn = (const float*)d_in[8];
  const float*     b_ffnn = (const float*)d_in[9];
  float*           out    = (float*)d_out;

  const int n = in_sizes[0] / D_IN;   // 100000
  const int e = in_sizes[1];          // 1000000

  char*  ws  = (char*)d_ws;
  float* h   = (float*)ws;                                            // n*32 f32
  float* agg = (float*)(ws + (size_t)n * D_H * sizeof(float));        // n*32 f32
  int*   deg = (int*)  (ws + 2 * (size_t)n * D_H * sizeof(float));    // n i32

  const int gRows = (n + 127) / 128;                 // 128 rows per 8-wave block
  k_input_proj<<<gRows, 256, 0, stream>>>(feat, W_in, b_in, h, n);

  const int initN = n * D_H;
  k_init<<<(initN + 255) / 256, 256, 0, stream>>>(agg, deg, n);

  const long long tot = 2LL * e;
  k_deg<<<(int)((tot + 255) / 256), 256, 0, stream>>>(src, dst, deg, e);

  const int gEdge = (int)((tot + 127) / 128);        // 128 endpoints per block
  k_nbr<<<gEdge, 256, 0, stream>>>(h, src, dst, W_nbr, b_nbr, b_n1, agg, e);

  k_ffnn<<<gRows, 256, 0, stream>>>(h, agg, deg, W_ffnn, b_ffnn, out, n);
}